// MMNL_loss_37168646980391
// MI455X (gfx1250) — compile-verified
//
#include <hip/hip_runtime.h>
#include <cstdint>

// MMNL loss: B=65536 rows x L=100 items, K=5 mixture components.
// Memory-bound gather workload (~210 MB once-touched => ~9us floor @ 23.3 TB/s).
// CDNA5 path used: TDM tensor_load_to_lds to stage each row's contiguous index
// tile into LDS, double-buffered across 2 rows per wave via TENSORcnt ordering.

#define L_LEN 100
#define K_LEN 5
#define WAVES_PER_BLOCK 8
#define ROWS_PER_WAVE 2
#define IDX_SLOT_INTS 128  // 512 B per row slot (100 ints used)

typedef unsigned int u32x4 __attribute__((ext_vector_type(4)));
typedef int          i32x8 __attribute__((ext_vector_type(8)));
typedef int          i32x4 __attribute__((ext_vector_type(4)));

// Issue one TDM descriptor: copy L_LEN contiguous i32 from global -> LDS.
// D# layout per CDNA5 ISA (§8.3/§8.4).
// 6-arg builtin form (amdgpu-toolchain / clang-23 lane).
__device__ __forceinline__ void tdm_load_row(const int* gsrc, unsigned int lds_byte_off) {
  unsigned long long ga = (unsigned long long)(uintptr_t)gsrc;
  u32x4 g0 = {
    1u,                                                   // count=1, user mode
    lds_byte_off,                                         // lds_addr (bytes)
    (unsigned int)ga,                                     // global_addr[31:0]
    ((unsigned int)((ga >> 32) & 0x1FFFFFFull)) | (2u << 30) // addr[56:32] | type=2
  };
  i32x8 g1 = {
    (2 << 16),            // data_size = 4 bytes; no multicast/pad/iterate
    (int)(L_LEN << 16),   // tensor_dim0 = 100  (bits 79:48)
    (1 << 16),            // tensor_dim1 = 1    (bits 111:80)
    (int)(L_LEN << 16),   // tile_dim0  = 100   (bits 127:112)
    1,                    // tile_dim1  = 1     (bits 143:128)
    L_LEN,                // tensor_dim0_stride (bits 207:160), low 32
    0, 0
  };
  i32x4 gz4 = { 0, 0, 0, 0 };
  i32x8 gz8 = { 0, 0, 0, 0, 0, 0, 0, 0 };
  __builtin_amdgcn_tensor_load_to_lds(g0, g1, gz4, gz4, gz8, 0);
}

// Per-row work: 12 reductions over L, wave32 butterfly, final scalar.
__device__ __forceinline__ float row_result(const int* lds_slot, int lane,
                                            const float* __restrict__ x,
                                            const float* __restrict__ y,
                                            const float* __restrict__ z,
                                            const float* __restrict__ alpha,
                                            size_t N) {
  float acc[2 + 2 * K_LEN];
#pragma unroll
  for (int t = 0; t < 2 + 2 * K_LEN; ++t) acc[t] = 0.f;

  if (lane < L_LEN / 4) {                       // 25 active lanes, 4 items each
    const int4 iv = ((const int4*)lds_slot)[lane];
    const int ids[4] = { iv.x, iv.y, iv.z, iv.w };
#pragma unroll
    for (int j = 0; j < 4; ++j) {
      const int i = ids[j];
      const float xi = x[i];
      const float yi = y[i];
      acc[0] += __expf(xi);                     // sum exp(x_S)
      acc[1] += xi * yi;                        // dot(x_S, y_S)
#pragma unroll
      for (int k = 0; k < K_LEN; ++k) {
        const float e = __expf(z[(size_t)k * N + (size_t)i]);
        acc[2 + k]         += e;                // sum exp(z_S)
        acc[2 + K_LEN + k] += e * yi;           // sum exp(z_S)*y_S
      }
    }
  }

  // wave32 butterfly reduction of all 12 accumulators
#pragma unroll
  for (int off = 16; off >= 1; off >>= 1) {
#pragma unroll
    for (int t = 0; t < 2 + 2 * K_LEN; ++t)
      acc[t] += __shfl_xor(acc[t], off, 32);
  }

  float g = 0.f;
#pragma unroll
  for (int k = 0; k < K_LEN; ++k)
    g += alpha[k] * (acc[2 + K_LEN + k] / acc[2 + k]);   // alpha . comp

  const float num = __expf(acc[1]);
  const float den = acc[0] * g;
  return num / den;
}

__global__ __launch_bounds__(WAVES_PER_BLOCK * 32)
void mmnl_rows_kernel(const float* __restrict__ x, const float* __restrict__ y,
                      const float* __restrict__ z, const float* __restrict__ alpha,
                      const int* __restrict__ idx, float* __restrict__ row_out,
                      int B, int Ni) {
  __shared__ int lds_idx[WAVES_PER_BLOCK * ROWS_PER_WAVE * IDX_SLOT_INTS];
  const size_t N = (size_t)Ni;
  const int wave = threadIdx.x >> 5;
  const int lane = threadIdx.x & 31;
  const int row0 = (blockIdx.x * WAVES_PER_BLOCK + wave) * ROWS_PER_WAVE;
  if (row0 >= B) return;                       // uniform per wave; no barriers used
  const int row1 = row0 + 1;
  const bool two = (row1 < B);

  int* slot0 = lds_idx + (wave * ROWS_PER_WAVE) * IDX_SLOT_INTS;
  int* slot1 = slot0 + IDX_SLOT_INTS;

  // Kick off both rows' index DMA; TENSORcnt is in-order per wave.
  tdm_load_row(idx + (size_t)row0 * L_LEN, (unsigned int)(uintptr_t)slot0);
  if (two) {
    tdm_load_row(idx + (size_t)row1 * L_LEN, (unsigned int)(uintptr_t)slot1);
    __builtin_amdgcn_s_wait_tensorcnt(1);      // first transfer complete
  } else {
    __builtin_amdgcn_s_wait_tensorcnt(0);
  }

  const float r0 = row_result(slot0, lane, x, y, z, alpha, N);
  if (lane == 0) row_out[row0] = r0;

  if (two) {
    __builtin_amdgcn_s_wait_tensorcnt(0);      // second transfer complete
    const float r1 = row_result(slot1, lane, x, y, z, alpha, N);
    if (lane == 0) row_out[row1] = r1;
  }
}

// Deterministic fixed-order reduction of B per-row ratios -> -mean.
__global__ __launch_bounds__(1024)
void reduce_neg_mean_kernel(const float* __restrict__ v, float* __restrict__ out, int n) {
  __shared__ float sm[1024];
  float s = 0.f;
  for (int i = threadIdx.x; i < n; i += 1024) s += v[i];
  sm[threadIdx.x] = s;
  __syncthreads();
  for (int off = 512; off > 0; off >>= 1) {
    if ((int)threadIdx.x < off) sm[threadIdx.x] += sm[threadIdx.x + off];
    __syncthreads();
  }
  if (threadIdx.x == 0) out[0] = -sm[0] / (float)n;
}

extern "C" void kernel_launch(void* const* d_in, const int* in_sizes, int n_in,
                              void* d_out, int out_size, void* d_ws, size_t ws_size,
                              hipStream_t stream) {
  const float* x     = (const float*)d_in[0];
  const float* y     = (const float*)d_in[1];
  const float* z     = (const float*)d_in[2];
  const float* alpha = (const float*)d_in[3];
  const int*   idx   = (const int*)d_in[4];

  const int N = in_sizes[0];
  const int B = in_sizes[4] / L_LEN;

  float* rows = (float*)d_ws;                  // B floats of scratch

  const int rows_per_block = WAVES_PER_BLOCK * ROWS_PER_WAVE;
  const int blocks = (B + rows_per_block - 1) / rows_per_block;
  mmnl_rows_kernel<<<blocks, WAVES_PER_BLOCK * 32, 0, stream>>>(
      x, y, z, alpha, idx, rows, B, N);
  reduce_neg_mean_kernel<<<1, 1024, 0, stream>>>(rows, (float*)d_out, B);
}